// QuantumClassifier_27814208209132
// MI455X (gfx1250) — compile-verified
//
#include <hip/hip_runtime.h>
#include <hip/hip_bf16.h>
#include <math.h>
#include <stdint.h>

typedef __attribute__((ext_vector_type(2))) float v2f;
typedef __attribute__((ext_vector_type(8))) float v8f;

#define DIM     4096
#define BATCH   128
#define NQ      12
#define NLAYERS 4
#define NCLASS  10

// ---------------- Kernel 1: X = GELU(A @ W^T + bias) -----------------------
// C[m][n] = sum_k A[m][k] * W[n][k];  M=128, N=4096, K=4096, all fp32.
// WG = 256 threads (8 wave32). Output tile per WG: 128 (all M) x 64 (N).
// K staged through double-buffered LDS via GLOBAL_LOAD_ASYNC_TO_LDS_B128.
#define KT  32
#define NWG 64
#define LDA (KT + 4) /* 36-float padded LDS row: 16B-aligned, conflict-skewed */

__device__ __forceinline__ float gelu_exact(float x) {
  return 0.5f * x * (1.0f + erff(x * 0.70710678118654752f));
}

__device__ __forceinline__ void async_ld_b128(unsigned lds_off, const void* gptr) {
  unsigned long long ga = (unsigned long long)gptr;
  asm volatile("global_load_async_to_lds_b128 %0, %1, off"
               :: "v"(lds_off), "v"(ga) : "memory");
}

__device__ __forceinline__ void wait_async0() {
#if __has_builtin(__builtin_amdgcn_s_wait_asynccnt)
  __builtin_amdgcn_s_wait_asynccnt(0);
#else
  asm volatile("s_wait_asynccnt 0x0" ::: "memory");
#endif
}

__device__ __forceinline__ unsigned lds_addr(const void* p) {
  // Generic LDS pointer = {shared aperture, 32-bit LDS byte offset}.
  return (unsigned)(uintptr_t)p;
}

__device__ __forceinline__ v8f wmma_f32(v2f a, v2f b, v8f c) {
  return __builtin_amdgcn_wmma_f32_16x16x4_f32(false, a, false, b, (short)0, c,
                                               false, false);
}

__global__ __launch_bounds__(256)
void pre_gemm_gelu(const float* __restrict__ A, const float* __restrict__ W,
                   const float* __restrict__ bias, float* __restrict__ X) {
  __shared__ float As[2][BATCH * LDA];
  __shared__ float Ws[2][NWG * LDA];
  const int t     = threadIdx.x;
  const int wave  = t >> 5;
  const int lane  = t & 31;
  const int half  = lane >> 4;     // 0: lanes 0-15, 1: lanes 16-31
  const int r     = lane & 15;
  const int nbase = blockIdx.x * NWG;
  const int m0    = wave * 16;

  // Per-thread staging coordinates (A: 4 x b128, W: 2 x b128 per K-tile).
  const int arow0 = (t + 0 * 256) >> 3, ac0 = ((t + 0 * 256) & 7) << 2;
  const int arow1 = (t + 1 * 256) >> 3, ac1 = ((t + 1 * 256) & 7) << 2;
  const int arow2 = (t + 2 * 256) >> 3, ac2 = ((t + 2 * 256) & 7) << 2;
  const int arow3 = (t + 3 * 256) >> 3, ac3 = ((t + 3 * 256) & 7) << 2;
  const int wrow0 = (t + 0 * 256) >> 3, wc0 = ((t + 0 * 256) & 7) << 2;
  const int wrow1 = (t + 1 * 256) >> 3, wc1 = ((t + 1 * 256) & 7) << 2;

  v8f acc[4];
#pragma unroll
  for (int j = 0; j < 4; ++j) acc[j] = (v8f)0.0f;

  // ---- stage K-tile k0 into buffer `b` (async, no wait here) ----
  auto stage = [&](int k0, int b) {
    async_ld_b128(lds_addr(&As[b][arow0 * LDA + ac0]), A + (size_t)arow0 * DIM + k0 + ac0);
    async_ld_b128(lds_addr(&As[b][arow1 * LDA + ac1]), A + (size_t)arow1 * DIM + k0 + ac1);
    async_ld_b128(lds_addr(&As[b][arow2 * LDA + ac2]), A + (size_t)arow2 * DIM + k0 + ac2);
    async_ld_b128(lds_addr(&As[b][arow3 * LDA + ac3]), A + (size_t)arow3 * DIM + k0 + ac3);
    async_ld_b128(lds_addr(&Ws[b][wrow0 * LDA + wc0]), W + (size_t)(nbase + wrow0) * DIM + k0 + wc0);
    async_ld_b128(lds_addr(&Ws[b][wrow1 * LDA + wc1]), W + (size_t)(nbase + wrow1) * DIM + k0 + wc1);
  };

  // ---- 4 WMMAs per K-step, fragments software-pipelined one step ahead ----
  auto compute = [&](int b) {
    const float* arow = &As[b][(m0 + r) * LDA + 2 * half];
    const float* wrow = &Ws[b][r * LDA + 2 * half];
    v2f a  = *(const v2f*)(arow);
    v2f b0 = *(const v2f*)(wrow);
    v2f b1 = *(const v2f*)(wrow + 16 * LDA);
    v2f b2 = *(const v2f*)(wrow + 32 * LDA);
    v2f b3 = *(const v2f*)(wrow + 48 * LDA);
#pragma unroll
    for (int kk = 4; kk <= KT; kk += 4) {
      v2f an = a, bn0 = b0, bn1 = b1, bn2 = b2, bn3 = b3;
      if (kk < KT) {
        an  = *(const v2f*)(arow + kk);
        bn0 = *(const v2f*)(wrow + kk);
        bn1 = *(const v2f*)(wrow + 16 * LDA + kk);
        bn2 = *(const v2f*)(wrow + 32 * LDA + kk);
        bn3 = *(const v2f*)(wrow + 48 * LDA + kk);
      }
      acc[0] = wmma_f32(a, b0, acc[0]);
      acc[1] = wmma_f32(a, b1, acc[1]);
      acc[2] = wmma_f32(a, b2, acc[2]);
      acc[3] = wmma_f32(a, b3, acc[3]);
      a = an; b0 = bn0; b1 = bn1; b2 = bn2; b3 = bn3;
    }
  };

  stage(0, 0);
  wait_async0();
  __syncthreads();

  int buf = 0;
  for (int k0 = KT; k0 < DIM; k0 += KT) {
    stage(k0, buf ^ 1);   // prefetch next tile while computing current
    compute(buf);
    wait_async0();        // this wave's async writes to buf^1 complete
    __syncthreads();      // all waves: buf^1 ready, buf fully consumed
    buf ^= 1;
  }
  compute(buf);

  // Epilogue: C/D layout -> VGPR v: lanes 0-15 = (M=v, N=lane), 16-31 = (M=8+v)
#pragma unroll
  for (int j = 0; j < 4; ++j) {
    int n    = nbase + j * 16 + r;
    float bn = bias[n];
#pragma unroll
    for (int v = 0; v < 8; ++v) {
      int m = m0 + half * 8 + v;
      X[m * DIM + n] = gelu_exact(acc[j][v] + bn);
    }
  }
}

// ---------------- Kernel 2: LDS-resident statevector simulation ------------
// One workgroup per batch element; psi (4096 complex64 = 32 KB) lives in LDS.
__device__ __forceinline__ float2 cmul(float2 a, float2 b) {
  return make_float2(a.x * b.x - a.y * b.y, a.x * b.y + a.y * b.x);
}

__global__ __launch_bounds__(256)
void quantum_sim(const float* __restrict__ X, const float* __restrict__ qw,
                 const float* __restrict__ Wfc, const float* __restrict__ bfc,
                 float* __restrict__ out) {
  __shared__ float2 psi[DIM];
  __shared__ float2 umat[NLAYERS * NQ * 4];  // u00,u01,u10,u11 per (layer,qubit)
  __shared__ float  red[256];
  __shared__ float  zsh[NQ];

  const int t = threadIdx.x;
  const int b = blockIdx.x;

  // Precompute the 48 Rot(phi,theta,omega) = RZ(omega)RY(theta)RZ(phi) matrices.
  if (t < NLAYERS * NQ) {
    float phi = qw[t * 3 + 0], theta = qw[t * 3 + 1], omega = qw[t * 3 + 2];
    float c = cosf(0.5f * theta), s = sinf(0.5f * theta);
    float ap = -0.5f * (phi + omega);   // ep = e^{-i(phi+omega)/2}
    float am =  0.5f * (phi - omega);   // em = e^{ i(phi-omega)/2}
    float2 ep = make_float2(cosf(ap), sinf(ap));
    float2 em = make_float2(cosf(am), sinf(am));
    umat[t * 4 + 0] = make_float2( ep.x * c,  ep.y * c);   // ep*c
    umat[t * 4 + 1] = make_float2(-em.x * s, -em.y * s);   // -em*s
    umat[t * 4 + 2] = make_float2( em.x * s, -em.y * s);   // conj(em)*s
    umat[t * 4 + 3] = make_float2( ep.x * c, -ep.y * c);   // conj(ep)*c
  }
  if (t < NQ) zsh[t] = 0.0f;

  // Amplitude embedding: load row, L2-normalize.
  float xs[16];
  float ss = 0.0f;
#pragma unroll
  for (int i = 0; i < 16; ++i) {
    float xv = X[(size_t)b * DIM + t + i * 256];
    xs[i] = xv;
    ss += xv * xv;
  }
  red[t] = ss;
  __syncthreads();
  for (int off = 128; off > 0; off >>= 1) {
    if (t < off) red[t] += red[t + off];
    __syncthreads();
  }
  float inv = 1.0f / sqrtf(red[0]);
#pragma unroll
  for (int i = 0; i < 16; ++i)
    psi[t + i * 256] = make_float2(xs[i] * inv, 0.0f);
  __syncthreads();

  // 4 reps of the (4-layer) strongly-entangling block. Wire 0 = MSB.
  for (int rep = 0; rep < NLAYERS; ++rep) {
    for (int l = 0; l < NLAYERS; ++l) {
      for (int q = 0; q < NQ; ++q) {
        const float2 u00 = umat[(l * NQ + q) * 4 + 0];
        const float2 u01 = umat[(l * NQ + q) * 4 + 1];
        const float2 u10 = umat[(l * NQ + q) * 4 + 2];
        const float2 u11 = umat[(l * NQ + q) * 4 + 3];
        const int stride = 1 << (NQ - 1 - q);
#pragma unroll 2
        for (int p = t; p < DIM / 2; p += 256) {
          int i0 = ((p & ~(stride - 1)) << 1) | (p & (stride - 1));
          int i1 = i0 + stride;
          float2 a0 = psi[i0], a1 = psi[i1];
          float2 r0 = cmul(u00, a0), r1 = cmul(u10, a0);
          float2 s0 = cmul(u01, a1), s1 = cmul(u11, a1);
          psi[i0] = make_float2(r0.x + s0.x, r0.y + s0.y);
          psi[i1] = make_float2(r1.x + s1.x, r1.y + s1.y);
        }
        __syncthreads();
      }
      const int rr = l % (NQ - 1) + 1;
      for (int q = 0; q < NQ; ++q) {
        const int tq = (q + rr) % NQ;
        const int pc = NQ - 1 - q, pt = NQ - 1 - tq;
        const int hi = pc > pt ? pc : pt, lo = pc > pt ? pt : pc;
        const int cb = 1 << pc, tb = 1 << pt;
#pragma unroll 2
        for (int p = t; p < DIM / 4; p += 256) {
          int v = p;
          int lm = v & ((1 << lo) - 1);
          v = ((v >> lo) << (lo + 1)) | lm;        // insert 0 at lo
          int mm = v & ((1 << hi) - 1);
          v = ((v >> hi) << (hi + 1)) | mm;        // insert 0 at hi
          int i0 = v | cb;                         // control=1, target=0
          int i1 = i0 | tb;                        // control=1, target=1
          float2 tmp = psi[i0];
          psi[i0] = psi[i1];
          psi[i1] = tmp;
        }
        __syncthreads();
      }
    }
  }

  // <Z_i> = sum_s |psi_s|^2 * (1 - 2*bit_i(s)), then the 12->10 fc layer.
  float zloc[NQ];
#pragma unroll
  for (int w = 0; w < NQ; ++w) zloc[w] = 0.0f;
#pragma unroll
  for (int i0 = 0; i0 < 16; ++i0) {
    int i = t + i0 * 256;
    float2 a = psi[i];
    float p = a.x * a.x + a.y * a.y;
#pragma unroll
    for (int w = 0; w < NQ; ++w)
      zloc[w] += ((i >> (NQ - 1 - w)) & 1) ? -p : p;
  }
#pragma unroll
  for (int w = 0; w < NQ; ++w) atomicAdd(&zsh[w], zloc[w]);
  __syncthreads();
  if (t < NCLASS) {
    float acc = bfc[t];
#pragma unroll
    for (int w = 0; w < NQ; ++w) acc += zsh[w] * Wfc[t * NQ + w];
    out[b * NCLASS + t] = acc;
  }
}

// ---------------------------------------------------------------------------
extern "C" void kernel_launch(void* const* d_in, const int* in_sizes, int n_in,
                              void* d_out, int out_size, void* d_ws, size_t ws_size,
                              hipStream_t stream) {
  const float* inputs    = (const float*)d_in[0];   // (128, 4096)
  const float* W_pre     = (const float*)d_in[1];   // (4096, 4096)
  const float* b_pre     = (const float*)d_in[2];   // (4096,)
  const float* q_weights = (const float*)d_in[3];   // (4, 12, 3)
  const float* W_fc      = (const float*)d_in[4];   // (10, 12)
  const float* b_fc      = (const float*)d_in[5];   // (10,)
  float* out = (float*)d_out;                       // (128, 10)
  float* X   = (float*)d_ws;                        // (128, 4096) scratch

  pre_gemm_gelu<<<DIM / NWG, 256, 0, stream>>>(inputs, W_pre, b_pre, X);
  quantum_sim<<<BATCH, 256, 0, stream>>>(X, q_weights, W_fc, b_fc, out);
}